// GnnLayer_27058293965311
// MI455X (gfx1250) — compile-verified
//
#include <hip/hip_runtime.h>

// ---------------------------------------------------------------------------
// GNN layer for MI455X (gfx1250), fp32 end-to-end.
//   out[m] = relu(Z[m]@Wr + mean_masked(gather(Z[m]@Wnr, nb[m])))
// Phase 1: fused dual GEMM (R and B share Z fragments), fp32 matrix pipe
//          (V_WMMA_F32_16X16X4_F32).
//   - Z tiles: GLOBAL_LOAD_ASYNC_TO_LDS_B128 (ASYNCcnt), double-buffered.
//   - W tiles: global->VGPR->LDS with pair-row interleave so every WMMA
//     B-fragment is a single aligned ds_load_b64 (no VGPR marshalling);
//     stores issued after the WMMA loop to overlap load latency.
// Phase 2: masked neighbor gather + mean + add + ReLU, float4 (B128) loads.
// ---------------------------------------------------------------------------

typedef __attribute__((ext_vector_type(2))) float v2f;
typedef __attribute__((ext_vector_type(8))) float v8f;

#define NROWS 50000
#define DK    256      // inner dim
#define FF    256      // output features
#define KNB   10       // neighbors

// GEMM tiling: block computes 64 rows x 64 cols of BOTH R and B.
#define BM 64
#define BN 64
#define KC 32          // K chunk staged in LDS
#define LDZ 36         // Z tile row stride (floats): gcd(36,64)=4 -> conflict-free A reads
#define LDW2 160       // W pair-row stride (floats): 64 cols * 2 interleaved + 32 pad
                       // (160 % 64 == 32 -> lane-halves hit disjoint bank sets)

// ---- CDNA5 async global->LDS copy (ASYNCcnt), GVS addressing ---------------
__device__ __forceinline__ unsigned ldsOffset(const void* p) {
    // generic shared-aperture address: bits[63:32] = aperture, bits[31:0] = LDS offset
    return (unsigned)(uintptr_t)p;
}
__device__ __forceinline__ void asyncLoadB128(unsigned ldsaddr, unsigned gbyteoff, const void* base) {
    asm volatile("global_load_async_to_lds_b128 %0, %1, %2"
                 :: "v"(ldsaddr), "v"(gbyteoff), "s"(base) : "memory");
}
__device__ __forceinline__ void waitAsync0() {
    asm volatile("s_wait_asynccnt 0" ::: "memory");
}

// Z tile 64x32 floats staged fully async (512 B128s / 256 threads).
__device__ __forceinline__ void stageZ(const float* Z, float* sZbuf,
                                       int rowBase, int k0, int tid)
{
    #pragma unroll
    for (int p = 0; p < 2; ++p) {
        const int idx = p * 256 + tid;
        const int r = idx >> 3;              // 0..63
        const int c = (idx & 7) * 4;         // 0..28
        int grow = rowBase + r;
        grow = grow < NROWS ? grow : NROWS - 1;           // branchless clamp
        const unsigned goff = (unsigned)(grow * DK + k0 + c) * 4u;
        asyncLoadB128(ldsOffset(sZbuf + r * LDZ + c), goff, Z);
    }
}

// W tiles 32x64 floats each: global float4 loads (issue early) ...
__device__ __forceinline__ void loadW(const float* Wr, const float* Wnr,
                                      int colBase, int k0, int tid,
                                      float4 wr[2], float4 wn[2])
{
    #pragma unroll
    for (int p = 0; p < 2; ++p) {
        const int idx = p * 256 + tid;
        const int r = idx >> 4;              // 0..31
        const int c = (idx & 15) * 4;        // 0..60
        wr[p] = *(const float4*)(Wr  + (size_t)(k0 + r) * FF + colBase + c);
        wn[p] = *(const float4*)(Wnr + (size_t)(k0 + r) * FF + colBase + c);
    }
}
// ... then scatter pair-interleaved: element (k,n) -> sW[(k>>1)*LDW2 + 2n + (k&1)]
__device__ __forceinline__ void storeW(float* sWrBuf, float* sWnBuf, int tid,
                                       const float4 wr[2], const float4 wn[2])
{
    #pragma unroll
    for (int p = 0; p < 2; ++p) {
        const int idx = p * 256 + tid;
        const int r = idx >> 4;
        const int c = (idx & 15) * 4;
        float* d0 = sWrBuf + (r >> 1) * LDW2 + c * 2 + (r & 1);
        d0[0] = wr[p].x; d0[2] = wr[p].y; d0[4] = wr[p].z; d0[6] = wr[p].w;
        float* d1 = sWnBuf + (r >> 1) * LDW2 + c * 2 + (r & 1);
        d1[0] = wn[p].x; d1[2] = wn[p].y; d1[4] = wn[p].z; d1[6] = wn[p].w;
    }
}

__global__ __launch_bounds__(256)
void gemm_dual_wmma(const float* __restrict__ Z,
                    const float* __restrict__ Wr,
                    const float* __restrict__ Wnr,
                    float* __restrict__ R,      // -> d_out region (residues signal)
                    float* __restrict__ Bout)   // -> workspace   (neighbor base signal)
{
    __shared__ float sZ[2][BM * LDZ];
    __shared__ float sWr[2][(KC / 2) * LDW2];
    __shared__ float sWn[2][(KC / 2) * LDW2];

    const int tid  = threadIdx.x;
    const int lane = tid & 31;
    const int wave = tid >> 5;                 // 8 waves
    const int rowBase = blockIdx.x * BM;
    const int colBase = blockIdx.y * BN;

    const int mrow  = (wave & 3) * 16;         // wave's 16-row sub-tile
    const int ncolW = (wave >> 2) * 32;        // wave's 32-col span (2 sub-tiles)

    v8f cR0 = {}, cR1 = {}, cB0 = {}, cB1 = {};

    const int halfsel = lane >> 4;             // 0: lanes 0-15, 1: lanes 16-31
    const int l15     = lane & 15;
    const int aRow    = (mrow + l15) * LDZ + halfsel * 2;   // + kk -> A pair (K-contiguous)
    const int nc2     = ncolW * 2 + l15 * 2;                // interleaved column offset

    // prologue: stage chunk 0
    {
        float4 wr[2], wn[2];
        stageZ(Z, sZ[0], rowBase, 0, tid);
        loadW(Wr, Wnr, colBase, 0, tid, wr, wn);
        storeW(sWr[0], sWn[0], tid, wr, wn);
        waitAsync0();
        __syncthreads();
    }

    int buf = 0;
    for (int k0 = 0; k0 < DK; k0 += KC) {
        const bool hasNext = (k0 + KC < DK);   // block-uniform
        float4 wr[2], wn[2];
        if (hasNext) {
            stageZ(Z, sZ[buf ^ 1], rowBase, k0 + KC, tid);     // async, no VGPRs
            loadW(Wr, Wnr, colBase, k0 + KC, tid, wr, wn);     // latency hidden by WMMAs
        }

        const float* cZ  = sZ[buf];
        const float* cWr = sWr[buf];
        const float* cWn = sWn[buf];

        #pragma unroll
        for (int kk = 0; kk < KC; kk += 4) {
            // A: lanes 0-15 K={kk,kk+1}, lanes 16-31 K={kk+2,kk+3}; one b64 each
            const v2f a = *(const v2f*)(cZ + aRow + kk);
            // B: VGPR0 rows {ka}, VGPR1 rows {ka+1}; pair-row = kk/2 + halfsel
            const float* pwr = cWr + ((kk >> 1) + halfsel) * LDW2 + nc2;
            const float* pwn = cWn + ((kk >> 1) + halfsel) * LDW2 + nc2;
            const v2f br0 = *(const v2f*)(pwr);
            const v2f br1 = *(const v2f*)(pwr + 32);
            const v2f bn0 = *(const v2f*)(pwn);
            const v2f bn1 = *(const v2f*)(pwn + 32);
            cR0 = __builtin_amdgcn_wmma_f32_16x16x4_f32(false, a, false, br0, (short)0, cR0, false, false);
            cR1 = __builtin_amdgcn_wmma_f32_16x16x4_f32(false, a, false, br1, (short)0, cR1, false, false);
            cB0 = __builtin_amdgcn_wmma_f32_16x16x4_f32(false, a, false, bn0, (short)0, cB0, false, false);
            cB1 = __builtin_amdgcn_wmma_f32_16x16x4_f32(false, a, false, bn1, (short)0, cB1, false, false);
        }

        if (hasNext)
            storeW(sWr[buf ^ 1], sWn[buf ^ 1], tid, wr, wn);   // after compute: loads landed

        waitAsync0();          // my async Z writes into buf^1 have landed
        __syncthreads();       // everyone's writes landed; everyone done reading buf
        buf ^= 1;
    }

    // store: C/D layout — VGPR r: lanes 0-15 -> M=r, lanes 16-31 -> M=r+8
    const int rbase = rowBase + mrow + (halfsel ? 8 : 0);
    const int col0  = colBase + ncolW + l15;
    #pragma unroll
    for (int r = 0; r < 8; ++r) {
        const int row = rbase + r;
        if (row < NROWS) {
            R   [(size_t)row * FF + col0]      = cR0[r];
            R   [(size_t)row * FF + col0 + 16] = cR1[r];
            Bout[(size_t)row * FF + col0]      = cB0[r];
            Bout[(size_t)row * FF + col0 + 16] = cB1[r];
        }
    }
}

__global__ __launch_bounds__(256)
void gather_finalize(const float* __restrict__ Bmat,
                     const int*   __restrict__ nb,
                     float*       __restrict__ out)   // holds R in-place; rewritten with final
{
    const int grp = threadIdx.x >> 6;                // 4 row-groups of 64 lanes
    const int l64 = threadIdx.x & 63;
    const int row = blockIdx.x * 4 + grp;
    const int c   = l64 * 4;

    int idxs[KNB];
    #pragma unroll
    for (int k = 0; k < KNB; ++k) idxs[k] = nb[row * KNB + k];

    int cnt = 0;
    float4 acc = make_float4(0.f, 0.f, 0.f, 0.f);
    #pragma unroll
    for (int k = 0; k < KNB; ++k) {
        const int idx = idxs[k];
        if (idx > -1) {                               // uniform per row-group
            ++cnt;
            const float4 v = *(const float4*)(Bmat + (size_t)idx * FF + c);
            acc.x += v.x; acc.y += v.y; acc.z += v.z; acc.w += v.w;
        }
    }
    const float inv = 1.0f / (float)(cnt > 0 ? cnt : 1);
    float4* o = (float4*)(out + (size_t)row * FF + c);
    float4 r = *o;
    r.x = fmaxf(fmaf(acc.x, inv, r.x), 0.f);
    r.y = fmaxf(fmaf(acc.y, inv, r.y), 0.f);
    r.z = fmaxf(fmaf(acc.z, inv, r.z), 0.f);
    r.w = fmaxf(fmaf(acc.w, inv, r.w), 0.f);
    *o = r;
}

extern "C" void kernel_launch(void* const* d_in, const int* in_sizes, int n_in,
                              void* d_out, int out_size, void* d_ws, size_t ws_size,
                              hipStream_t stream)
{
    (void)in_sizes; (void)n_in; (void)out_size; (void)ws_size;

    const float* Z1  = (const float*)d_in[0];
    const float* Z2  = (const float*)d_in[1];
    const float* Wr  = (const float*)d_in[2];
    const float* Wnr = (const float*)d_in[3];
    const int*   n1  = (const int*)d_in[4];
    const int*   n2  = (const int*)d_in[5];

    float* out = (float*)d_out;                       // [2, N, F]
    float* B1  = (float*)d_ws;                        // [N, F] neighbor-base signal, matrix 1
    float* B2  = B1 + (size_t)NROWS * FF;             // [N, F] matrix 2

    const size_t matStride = (size_t)NROWS * FF;

    dim3 grid((NROWS + BM - 1) / BM, FF / BN, 1);     // 782 x 4
    gemm_dual_wmma<<<grid, 256, 0, stream>>>(Z1, Wr, Wnr, out,             B1);
    gemm_dual_wmma<<<grid, 256, 0, stream>>>(Z2, Wr, Wnr, out + matStride, B2);

    gather_finalize<<<NROWS / 4, 256, 0, stream>>>(B1, n1, out);
    gather_finalize<<<NROWS / 4, 256, 0, stream>>>(B2, n2, out + matStride);
}